// MambaBlock_9354438771056
// MI455X (gfx1250) — compile-verified
//
#include <hip/hip_runtime.h>
#include <hip/hip_bf16.h>
#include <math.h>

typedef __attribute__((ext_vector_type(2))) float v2f;
typedef __attribute__((ext_vector_type(8))) float v8f;

#define D_MODEL 1024
#define D_STATE 32
#define KERW    4
#define DT_RANK 64
#define BATCH   2
#define SEQLEN  1024
#define NTOK    (BATCH * SEQLEN)   // 2048 tokens

__device__ __forceinline__ float sigmoidf_(float v) { return 1.0f / (1.0f + __expf(-v)); }

// =====================================================================
// fp32 WMMA GEMM:  C[M,N] = act( A[M,K] @ B[N,K]^T + bias[N] )
//   A row-stride = K, B row-stride = K, C row-stride = N.
//   One wave computes a (16*MSUB) x (16*NSUB) tile via
//   V_WMMA_F32_16X16X4_F32.  MSUB/NSUB are compile-time so the inner
//   loop is straight-line (no EXEC-mask branches around WMMA).
//   Requires: M % (16*MSUB) == 0, N % (16*NSUB) == 0, K % 8 == 0.
//   ACT: 0 = none, 1 = softplus.
// =====================================================================
template<int MSUB, int NSUB, int ACT>
__global__ __launch_bounds__(128)
void wmma_gemm_nt(const float* __restrict__ A, const float* __restrict__ Bw,
                  const float* __restrict__ bias, float* __restrict__ C,
                  int M, int N, int K)
{
    const int lane   = threadIdx.x & 31;
    const int wave   = blockIdx.x * (blockDim.x >> 5) + (threadIdx.x >> 5);
    const int mtiles = M / (16 * MSUB);
    const int ntiles = N / (16 * NSUB);
    if (wave >= mtiles * ntiles) return;           // uniform per wave
    const int m0 = (wave % mtiles) * (16 * MSUB);
    const int n0 = (wave / mtiles) * (16 * NSUB);

    // Fragment addressing (16x4 f32 A / 4x16 f32 B):
    //   lanes 0-15: rows/cols 0-15 at K = {k, k+1};  lanes 16-31: same at K = {k+2, k+3}
    const int row16 = lane & 15;
    const int khalf = (lane >> 4) << 1;            // 0 or 2

    const float* Ap[MSUB];
    const float* Bp[NSUB];
    #pragma unroll
    for (int mi = 0; mi < MSUB; ++mi)
        Ap[mi] = A + (size_t)(m0 + mi * 16 + row16) * K + khalf;
    #pragma unroll
    for (int ni = 0; ni < NSUB; ++ni)
        Bp[ni] = Bw + (size_t)(n0 + ni * 16 + row16) * K + khalf;

    v8f acc[MSUB][NSUB] = {};

    #pragma unroll 2
    for (int k = 0; k < K; k += 4) {
        v2f af[MSUB];
        v2f bf[NSUB];
        #pragma unroll
        for (int mi = 0; mi < MSUB; ++mi) af[mi] = *(const v2f*)(Ap[mi] + k);
        #pragma unroll
        for (int ni = 0; ni < NSUB; ++ni) bf[ni] = *(const v2f*)(Bp[ni] + k);
        #pragma unroll
        for (int ni = 0; ni < NSUB; ++ni)
            #pragma unroll
            for (int mi = 0; mi < MSUB; ++mi)
                acc[mi][ni] = __builtin_amdgcn_wmma_f32_16x16x4_f32(
                    false, af[mi], false, bf[ni], (short)0, acc[mi][ni], false, false);
    }

    // C/D layout: VGPR i -> row (lane>=16 ? 8 : 0) + i, col lane&15
    const int crowbase = (lane >> 4) << 3;
    #pragma unroll
    for (int mi = 0; mi < MSUB; ++mi) {
        #pragma unroll
        for (int ni = 0; ni < NSUB; ++ni) {
            const int col  = n0 + ni * 16 + (lane & 15);
            const float ba = bias ? bias[col] : 0.0f;
            #pragma unroll
            for (int i = 0; i < 8; ++i) {
                float v = acc[mi][ni][i] + ba;
                if (ACT == 1) v = (v > 20.0f) ? v : log1pf(__expf(v));  // softplus
                C[(size_t)(m0 + mi * 16 + crowbase + i) * N + col] = v;
            }
        }
    }
}

// =====================================================================
// Depthwise causal conv1d (K=4) + SiLU.  Reads 'a' = az[:, :D_MODEL].
// =====================================================================
__global__ void conv_silu_kernel(const float* __restrict__ az, const float* __restrict__ cw,
                                 const float* __restrict__ cb, float* __restrict__ x)
{
    int idx = blockIdx.x * blockDim.x + threadIdx.x;
    if (idx >= NTOK * D_MODEL) return;
    const int d = idx & (D_MODEL - 1);
    const int t = idx >> 10;              // token = b*L + l
    const int l = t & (SEQLEN - 1);
    float acc = cb[d];
    #pragma unroll
    for (int j = 0; j < KERW; ++j) {
        int ll = l - (KERW - 1) + j;
        if (ll >= 0)
            acc += az[(size_t)(t - (KERW - 1) + j) * (2 * D_MODEL) + d] * cw[d * KERW + j];
    }
    x[idx] = acc * sigmoidf_(acc);
}

// =====================================================================
// Selective scan: one wave per (b,d); lane == state index (D_STATE==32).
// h kept in a register for all L steps; y-reduction over states via
// 5-step lane shuffle (wave32).
// =====================================================================
__global__ void scan_kernel(const float* __restrict__ delta, const float* __restrict__ Bv,
                            const float* __restrict__ Cv, const float* __restrict__ x,
                            const float* __restrict__ A_log, float* __restrict__ ysc)
{
    const int lane = threadIdx.x & 31;
    const int wave = blockIdx.x * (blockDim.x >> 5) + (threadIdx.x >> 5);
    if (wave >= BATCH * D_MODEL) return;
    const int b = wave >> 10;
    const int d = wave & (D_MODEL - 1);

    const float Acoef = -__expf(A_log[d * D_STATE + lane]);   // A = -exp(A_log)
    float h = 0.0f;
    const size_t tokBase = (size_t)b * SEQLEN;

    for (int l = 0; l < SEQLEN; ++l) {
        const size_t tok = tokBase + l;
        const float dt = delta[tok * D_MODEL + d];
        const float xt = x[tok * D_MODEL + d];
        const float Bt = Bv[tok * D_STATE + lane];
        const float Ct = Cv[tok * D_STATE + lane];
        h = __expf(Acoef * dt) * h + dt * Bt * xt;
        float p = h * Ct;
        #pragma unroll
        for (int off = 16; off > 0; off >>= 1) p += __shfl_xor(p, off, 32);
        if (lane == 0) ysc[tok * D_MODEL + d] = p;
    }
}

// =====================================================================
// y = (y_scan + D_skip*x) * silu(z)   (z = az[:, D_MODEL:])
// =====================================================================
__global__ void gate_kernel(const float* __restrict__ ysc, const float* __restrict__ x,
                            const float* __restrict__ az, const float* __restrict__ Dskip,
                            float* __restrict__ y)
{
    int idx = blockIdx.x * blockDim.x + threadIdx.x;
    if (idx >= NTOK * D_MODEL) return;
    const int d = idx & (D_MODEL - 1);
    const int t = idx >> 10;
    const float z = az[(size_t)t * (2 * D_MODEL) + D_MODEL + d];
    const float v = ysc[idx] + Dskip[d] * x[idx];
    y[idx] = v * z * sigmoidf_(z);
}

template<int MSUB, int NSUB, int ACT>
static inline void run_gemm(const float* A, const float* Bw, const float* bias, float* C,
                            int M, int N, int K, hipStream_t stream)
{
    int waves  = (M / (16 * MSUB)) * (N / (16 * NSUB));
    int blocks = (waves + 3) / 4;                 // 4 waves (128 threads) per block
    wmma_gemm_nt<MSUB, NSUB, ACT><<<dim3(blocks), dim3(128), 0, stream>>>(A, Bw, bias, C, M, N, K);
}

extern "C" void kernel_launch(void* const* d_in, const int* in_sizes, int n_in,
                              void* d_out, int out_size, void* d_ws, size_t ws_size,
                              hipStream_t stream)
{
    const float* seq    = (const float*)d_in[0];
    const float* W_in   = (const float*)d_in[1];
    const float* W_out  = (const float*)d_in[2];
    const float* conv_w = (const float*)d_in[3];
    const float* conv_b = (const float*)d_in[4];
    const float* W_sdt  = (const float*)d_in[5];
    const float* W_dtp  = (const float*)d_in[6];
    const float* b_dtp  = (const float*)d_in[7];
    const float* W_sB   = (const float*)d_in[8];
    const float* W_sC   = (const float*)d_in[9];
    const float* A_log  = (const float*)d_in[10];
    const float* D_skip = (const float*)d_in[11];
    float* out = (float*)d_out;

    // workspace layout (floats)
    float* ws    = (float*)d_ws;
    float* az    = ws;                                    // 2048 * 2048
    float* x     = az    + (size_t)NTOK * 2 * D_MODEL;    // 2048 * 1024
    float* t     = x     + (size_t)NTOK * D_MODEL;        // 2048 * 64
    float* delta = t     + (size_t)NTOK * DT_RANK;        // 2048 * 1024
    float* Bv    = delta + (size_t)NTOK * D_MODEL;        // 2048 * 32
    float* Cv    = Bv    + (size_t)NTOK * D_STATE;        // 2048 * 32
    float* ysc   = Cv    + (size_t)NTOK * D_STATE;        // 2048 * 1024
    float* y     = delta;                                 // reuse: delta consumed by scan

    // 1) az = seq @ W_in.T                  (M=2048, N=2048, K=1024)
    run_gemm<2, 4, 0>(seq, W_in, nullptr, az, NTOK, 2 * D_MODEL, D_MODEL, stream);
    // 2) x = silu(depthwise_conv(a) + b)
    conv_silu_kernel<<<(NTOK * D_MODEL + 255) / 256, 256, 0, stream>>>(az, conv_w, conv_b, x);
    // 3) t = x @ W_sdt.T                    (N=64)
    run_gemm<2, 4, 0>(x, W_sdt, nullptr, t, NTOK, DT_RANK, D_MODEL, stream);
    // 4) delta = softplus(t @ W_dtp.T + b)  (K=64)
    run_gemm<2, 4, 1>(t, W_dtp, b_dtp, delta, NTOK, D_MODEL, DT_RANK, stream);
    // 5) Bv, Cv = x @ W_sB.T, x @ W_sC.T    (N=32 -> NSUB=2)
    run_gemm<2, 2, 0>(x, W_sB, nullptr, Bv, NTOK, D_STATE, D_MODEL, stream);
    run_gemm<2, 2, 0>(x, W_sC, nullptr, Cv, NTOK, D_STATE, D_MODEL, stream);
    // 6) selective scan (2048 waves, 8 waves/block)
    scan_kernel<<<dim3(256), dim3(256), 0, stream>>>(delta, Bv, Cv, x, A_log, ysc);
    // 7) gate
    gate_kernel<<<(NTOK * D_MODEL + 255) / 256, 256, 0, stream>>>(ysc, x, az, D_skip, y);
    // 8) out = y @ W_out.T                  (M=2048, N=1024, K=1024)
    run_gemm<2, 4, 0>(y, W_out, nullptr, out, NTOK, D_MODEL, D_MODEL, stream);
}